// TopKGate_61564061221517
// MI455X (gfx1250) — compile-verified
//
#include <hip/hip_runtime.h>
#include <hip/hip_bf16.h>
#include <float.h>

typedef float v2f __attribute__((ext_vector_type(2)));
typedef float v8f __attribute__((ext_vector_type(8)));

constexpr int       S  = 65536;
constexpr int       Md = 1024;
constexpr int       E  = 64;
constexpr int       CAP = 2048;                       // TOP_K * ceil(S/E)
constexpr long long DISP_ELEMS = (long long)E * CAP * Md;  // 134217728

// ---------------------------------------------------------------------------
// Kernel 1: zero the dispatched region (harness does not re-poison between
// replays) and the per-expert top-1 counters in workspace.
// ---------------------------------------------------------------------------
__global__ __launch_bounds__(256) void moe_zero_kernel(float4* __restrict__ disp,
                                                       int* __restrict__ ce) {
  long long i = (long long)blockIdx.x * 256 + threadIdx.x;
  disp[i] = make_float4(0.f, 0.f, 0.f, 0.f);
  if (blockIdx.x == 0 && threadIdx.x < E) ce[threadIdx.x] = 0;
}

// ---------------------------------------------------------------------------
// Kernel 2: fused fp32 WMMA GEMM (logits = x @ wg^T) + softmax + top-2 +
// normalized gates + per-block me[e] partials + top-1 expert counts.
// Block = 128 threads (4 waves); each wave owns 16 tokens x all 64 experts:
// four 16x16 accumulators, so each A fragment (x) is loaded once and reused
// by 4 v_wmma_f32_16x16x4_f32 ops. K-loop: 1024/4 steps.
// ---------------------------------------------------------------------------
__global__ __launch_bounds__(128) void moe_gate_kernel(
    const float* __restrict__ x, const float* __restrict__ wg,
    float* __restrict__ gates_out, int* __restrict__ top_idx,
    float* __restrict__ me_part, int* __restrict__ ce) {
  __shared__ float lds_logits[64][E + 1];
  __shared__ float lds_max[64];
  __shared__ float lds_z[64];

  const int tid   = threadIdx.x;
  const int wave  = tid >> 5;
  const int lane  = tid & 31;
  const int row0  = blockIdx.x * 64 + wave * 16;  // this wave's 16 tokens
  const int mrow  = lane & 15;          // M (tokens) / N (experts) index
  const int khalf = (lane >> 4) << 1;   // lanes 0-15 -> K {0,1}; 16-31 -> K {2,3}

  const float* xrow = x  + (long long)(row0 + mrow) * Md + khalf;
  const float* wrow = wg + (long long)mrow * Md + khalf;  // col tile ct at +ct*16*Md

  v8f cacc[4] = {{}, {}, {}, {}};
#pragma unroll 4
  for (int k0 = 0; k0 < Md; k0 += 4) {
    v2f a = *(const v2f*)(xrow + k0);          // A: 16x4 f32 fragment (reused 4x)
#pragma unroll
    for (int ct = 0; ct < 4; ++ct) {
      v2f b = *(const v2f*)(wrow + ct * 16 * Md + k0);  // B: 4x16 frag (wg^T)
      cacc[ct] = __builtin_amdgcn_wmma_f32_16x16x4_f32(
          /*neg_a=*/false, a, /*neg_b=*/false, b,
          /*c_mod=*/(short)0, cacc[ct], /*reuse_a=*/false, /*reuse_b=*/false);
    }
  }

  // C/D layout: VGPR v holds M = v + 8*(lane>=16), N = lane&15
  const int mbase = wave * 16 + (lane >> 4) * 8;
  const int ncol  = lane & 15;
#pragma unroll
  for (int ct = 0; ct < 4; ++ct)
#pragma unroll
    for (int v = 0; v < 8; ++v)
      lds_logits[mbase + v][ct * 16 + ncol] = cacc[ct][v];
  __syncthreads();

  if (tid < 64) {  // one thread per token: softmax stats + top-2
    float mx = -FLT_MAX;
    for (int e = 0; e < E; ++e) mx = fmaxf(mx, lds_logits[tid][e]);
    float z = 0.f, v1 = -FLT_MAX, v2 = -FLT_MAX;
    int i1 = 0, i2 = 0;
    for (int e = 0; e < E; ++e) {
      float l = lds_logits[tid][e];
      z += __expf(l - mx);
      if (l > v1)      { v2 = v1; i2 = i1; v1 = l; i1 = e; }
      else if (l > v2) { v2 = l;  i2 = e; }
    }
    lds_max[tid] = mx;
    lds_z[tid]   = z;
    float g1 = __expf(v1 - mx) / z;
    float g2 = __expf(v2 - mx) / z;
    float denom = fmaxf(g1 + g2, 1.1920929e-07f);  // float32 eps
    long long s = (long long)blockIdx.x * 64 + tid;
    gates_out[s * 2 + 0] = g1 / denom;
    gates_out[s * 2 + 1] = g2 / denom;
    top_idx[s * 2 + 0] = i1;
    top_idx[s * 2 + 1] = i2;
    atomicAdd(&ce[i1], 1);  // integer atomic: deterministic
  }
  __syncthreads();

  if (tid < E) {  // per-block me[e] partial (fixed-order reduced later)
    float acc = 0.f;
    for (int t = 0; t < 64; ++t)
      acc += __expf(lds_logits[t][tid] - lds_max[t]) / lds_z[t];
    me_part[(long long)blockIdx.x * E + tid] = acc;
  }
}

// ---------------------------------------------------------------------------
// Kernel 3: l_loss = sum_e me[e]*ce[e] * E/(S*S), fixed-order reduction.
// ---------------------------------------------------------------------------
__global__ __launch_bounds__(64) void moe_loss_kernel(
    const float* __restrict__ me_part, const int* __restrict__ ce,
    float* __restrict__ lloss, int nblocks) {
  __shared__ float red[E];
  const int e = threadIdx.x;
  float me = 0.f;
  for (int b = 0; b < nblocks; ++b) me += me_part[(long long)b * E + e];
  red[e] = me * (float)ce[e];
  __syncthreads();
  for (int off = 32; off > 0; off >>= 1) {
    if (e < off) red[e] += red[e + off];
    __syncthreads();
  }
  if (e == 0) *lloss = red[0] * ((float)E / ((float)S * (float)S));
}

// ---------------------------------------------------------------------------
// Kernel 4: ordered per-(k,expert) scan over all tokens -> flat dispatch slot
// (or -1 when over capacity). Block b: k = b>>6, e = b&63. k=1 is offset by
// ce[e] (total top-1 count) per the reference's acc_base.
// ---------------------------------------------------------------------------
__global__ __launch_bounds__(256) void moe_scan_kernel(
    const int* __restrict__ top_idx, const int* __restrict__ ce,
    int* __restrict__ dst) {
  const int k   = blockIdx.x >> 6;
  const int e   = blockIdx.x & 63;
  const int tid = threadIdx.x;
  __shared__ int sc[256];
  int base = (k == 1) ? ce[e] : 0;
  for (int tile = 0; tile < S / 256; ++tile) {
    const int s = tile * 256 + tid;
    const int flag = (top_idx[(long long)s * 2 + k] == e) ? 1 : 0;
    sc[tid] = flag;
    __syncthreads();
    for (int off = 1; off < 256; off <<= 1) {  // Hillis-Steele inclusive scan
      int v = (tid >= off) ? sc[tid - off] : 0;
      __syncthreads();
      sc[tid] += v;
      __syncthreads();
    }
    const int incl  = sc[tid];
    const int total = sc[255];
    if (flag) {
      const int loc = base + incl - 1;
      dst[(long long)s * 2 + k] = (loc < CAP) ? (e * CAP + loc) : -1;
    }
    base += total;
    __syncthreads();
  }
}

// ---------------------------------------------------------------------------
// Kernel 5: scatter token rows into dispatched (each slot written at most
// once, so plain float4 stores — no atomics needed).
// ---------------------------------------------------------------------------
__global__ __launch_bounds__(256) void moe_scatter_kernel(
    const float* __restrict__ x, const int* __restrict__ dst,
    float* __restrict__ disp) {
  const int d = dst[blockIdx.x];
  if (d < 0) return;
  const long long s = blockIdx.x >> 1;
  const float4* src = (const float4*)(x + s * (long long)Md);
  float4* dp = (float4*)(disp + (long long)d * Md);
  dp[threadIdx.x] = src[threadIdx.x];
}

extern "C" void kernel_launch(void* const* d_in, const int* in_sizes, int n_in,
                              void* d_out, int out_size, void* d_ws, size_t ws_size,
                              hipStream_t stream) {
  const float* x  = (const float*)d_in[0];
  const float* wg = (const float*)d_in[1];

  float* out   = (float*)d_out;
  float* disp  = out;                       // [E, CAP, Md]
  float* lloss = out + DISP_ELEMS;          // scalar
  float* gates = out + DISP_ELEMS + 1;      // [S, 2]

  char* ws = (char*)d_ws;
  int*   top_idx = (int*)ws;                                  // S*2 ints
  int*   dstbuf  = (int*)(ws + (size_t)S * 2 * sizeof(int));  // S*2 ints
  int*   ce      = (int*)(ws + (size_t)S * 4 * sizeof(int));  // 64 ints (1KB pad)
  float* me_part = (float*)(ws + (size_t)S * 4 * sizeof(int) + 1024);  // (S/64)*E

  constexpr int NBLK = S / 64;  // 1024 gate blocks

  moe_zero_kernel<<<(int)(DISP_ELEMS / 4 / 256), 256, 0, stream>>>((float4*)disp, ce);
  moe_gate_kernel<<<NBLK, 128, 0, stream>>>(x, wg, gates, top_idx, me_part, ce);
  moe_loss_kernel<<<1, 64, 0, stream>>>(me_part, ce, lloss, NBLK);
  moe_scan_kernel<<<128, 256, 0, stream>>>(top_idx, ce, dstbuf);
  moe_scatter_kernel<<<S * 2, 256, 0, stream>>>(x, dstbuf, disp);
}